// GIN_63264868270409
// MI455X (gfx1250) — compile-verified
//
#include <hip/hip_runtime.h>
#include <math.h>

// ---------------- problem constants (match reference) ----------------
#define N_NODES  100000
#define N_EDGES  3200000
#define F_IN     128
#define H_DIM    128
#define C_CLS    10
#define N_GRAPHS 512
#define BN_EPS   1e-5f

typedef float v2f __attribute__((ext_vector_type(2)));
typedef float v8f __attribute__((ext_vector_type(8)));

// =====================================================================
// Persistent GEMM: Z[N,128] = op(A)[N,128] @ W[128,128] + bias
//   op(A) = useBN ? relu(A*sc[k]+sh[k]) : A     (BN+ReLU fused on load)
// V_WMMA_F32_16X16X4_F32, wave32. Block = 256 threads = 8 waves; wave w
// owns output cols [16w,16w+16) and keeps its whole 128x16 W stripe in
// registers (64 VGPR/lane) for the entire grid-stride loop over row
// tiles.  Steady-state inner loop: ds_load_2addr_b64 + v_wmma only.
// Per-output-column sum / sum^2 (for the next BatchNorm) accumulate in
// registers across all tiles -> one atomic pair per thread at the end.
// =====================================================================
__global__ __launch_bounds__(256)
void gin_gemm128(const float* __restrict__ A, const float* __restrict__ W,
                 const float* __restrict__ bias,
                 const float* __restrict__ sc, const float* __restrict__ sh,
                 int useBN,
                 float* __restrict__ Z,
                 float* __restrict__ statSum, float* __restrict__ statSqs,
                 int ntiles)
{
    __shared__ float As[16 * 128];          // 8 KB A tile
    const int tid   = threadIdx.x;
    const int wave  = tid >> 5;
    const int lane  = tid & 31;
    const int half  = lane >> 4;            // lanes 0-15: K0/K1 | lanes 16-31: K2/K3
    const int ln    = lane & 15;
    const int cbase = wave * 16;

    // ---- preload this wave's B stripe: W[k][cbase+ln], k-pairs per kt ----
    v2f bfrag[32];
    #pragma unroll
    for (int kt = 0; kt < 32; ++kt) {
        const int kb = kt * 4 + 2 * half;
        bfrag[kt].x = W[(size_t)(kb + 0) * 128 + cbase + ln];
        bfrag[kt].y = W[(size_t)(kb + 1) * 128 + cbase + ln];
    }
    const float bv = bias[cbase + ln];

    // ---- preload BN scale/shift for this thread's fixed 8 stage columns ----
    const int k0 = (tid * 8) & 127;
    float scr[8], shr[8];
    if (useBN) {
        #pragma unroll
        for (int i = 0; i < 8; ++i) { scr[i] = sc[k0 + i]; shr[i] = sh[k0 + i]; }
    }

    float csum = 0.0f, csqs = 0.0f;         // column stats (column fixed per lane)

    for (int t = blockIdx.x; t < ntiles; t += gridDim.x) {
        const int rb = t * 16;
        __syncthreads();                    // As consumed by previous iteration
        // stage A tile (8 consecutive floats per thread -> b128 loads)
        #pragma unroll
        for (int i = 0; i < 8; ++i) {
            const int idx = tid * 8 + i;    // 0..2047
            const int r   = idx >> 7;
            float v = A[(size_t)(rb + r) * 128 + (k0 + i)];
            if (useBN) v = fmaxf(fmaf(v, scr[i], shr[i]), 0.0f);
            As[idx] = v;
        }
        __syncthreads();

        v8f c;
        #pragma unroll
        for (int r = 0; r < 8; ++r) c[r] = bv;

        // K = 128 -> 32 x V_WMMA_F32_16X16X4_F32, A from LDS, B from VGPRs
        #pragma unroll
        for (int kt = 0; kt < 32; ++kt) {
            v2f a;
            a.x = As[ln * 128 + kt * 4 + 2 * half + 0];
            a.y = As[ln * 128 + kt * 4 + 2 * half + 1];
            c = __builtin_amdgcn_wmma_f32_16x16x4_f32(
                    /*neg_a=*/false, a, /*neg_b=*/false, bfrag[kt],
                    /*c_mod=*/(short)0, c, /*reuse_a=*/false, /*reuse_b=*/false);
        }

        // C/D layout: VGPR r -> row r (lanes 0-15) / r+8 (lanes 16-31);
        // every lane's 8 values share ONE output column (cbase+ln).
        #pragma unroll
        for (int r = 0; r < 8; ++r) {
            const float v = c[r];
            Z[(size_t)(rb + r + 8 * half) * 128 + cbase + ln] = v;
            csum += v;
            csqs += v * v;
        }
    }

    atomicAdd(&statSum[cbase + ln], csum);
    atomicAdd(&statSqs[cbase + ln], csqs);
}

// =====================================================================
// Fold batch stats into affine scale/shift:
//   sc = g * rsqrt(var + eps) ; sh = beta - mean * sc
// =====================================================================
__global__ void gin_stats(const float* __restrict__ sum, const float* __restrict__ sqs,
                          const float* __restrict__ g,  const float* __restrict__ be,
                          float* __restrict__ sc, float* __restrict__ sh, float invN)
{
    const int c = threadIdx.x;               // 128 threads
    const float m = sum[c] * invN;
    const float v = sqs[c] * invN - m * m;   // biased variance (training mode)
    const float s = g[c] * rsqrtf(v + BN_EPS);
    sc[c] = s;
    sh[c] = be[c] - m * s;
}

// =====================================================================
// Elementwise BN+ReLU with fused graph pooling.
//   v = relu(Z*sc+sh) ; optionally materialize H1 and Agg-init (H2);
//   pool[batch[node]] += v ; cnt[g] += 1 (once per node).
// =====================================================================
__global__ __launch_bounds__(256)
void gin_bnrelu_pool(const float* __restrict__ Z,
                     const float* __restrict__ sc, const float* __restrict__ sh,
                     const int* __restrict__ batch,
                     float* __restrict__ H1, float* __restrict__ H2,
                     float* __restrict__ pool, float* __restrict__ cnt)
{
    const int idx = blockIdx.x * 256 + threadIdx.x;   // < N_NODES*128 = 12.8M
    const int node = idx >> 7;
    const int col  = idx & 127;
    float v = fmaxf(fmaf(Z[idx], sc[col], sh[col]), 0.0f);
    if (H1) H1[idx] = v;
    if (H2) H2[idx] = v;                               // agg init (GIN eps=0 self term)
    const int g = batch[node];
    atomicAdd(&pool[g * 128 + col], v);
    if (cnt && col == 0) atomicAdd(&cnt[g], 1.0f);
}

// =====================================================================
// Edge scatter-add: Agg[dst[e]] += H[src[e]]  (wave32 per edge,
// float4 gather per lane + 4 global_atomic_add_f32)
// =====================================================================
__global__ __launch_bounds__(256)
void gin_scatter(const float* __restrict__ H,
                 const int* __restrict__ src, const int* __restrict__ dst,
                 float* __restrict__ Agg, int nE)
{
    const int wid  = (blockIdx.x * 256 + threadIdx.x) >> 5;
    const int lane = threadIdx.x & 31;
    if (wid >= nE) return;
    const int s = src[wid];
    const int d = dst[wid];
    const float4 v = ((const float4*)(H + (size_t)s * 128))[lane];
    float* ad = Agg + (size_t)d * 128 + lane * 4;
    atomicAdd(ad + 0, v.x);
    atomicAdd(ad + 1, v.y);
    atomicAdd(ad + 2, v.z);
    atomicAdd(ad + 3, v.w);
}

// =====================================================================
// Heads + log_softmax.  One block per graph.
//   logits = pool1 @ Wl0 + cnt*bl0 + pool2 @ Wl1 + bl1 ; log_softmax
// =====================================================================
__global__ __launch_bounds__(64)
void gin_heads(const float* __restrict__ pool1, const float* __restrict__ pool2,
               const float* __restrict__ cnt,
               const float* __restrict__ Wl0, const float* __restrict__ bl0,
               const float* __restrict__ Wl1, const float* __restrict__ bl1,
               float* __restrict__ out)
{
    const int g = blockIdx.x;
    const int t = threadIdx.x;
    __shared__ float logits[C_CLS];
    if (t < C_CLS) {
        float acc = cnt[g] * bl0[t] + bl1[t];
        const float* p1 = pool1 + g * 128;
        const float* p2 = pool2 + g * 128;
        #pragma unroll 4
        for (int k = 0; k < 128; ++k)
            acc += p1[k] * Wl0[k * C_CLS + t] + p2[k] * Wl1[k * C_CLS + t];
        logits[t] = acc;
    }
    __syncthreads();
    if (t == 0) {
        float mx = logits[0];
        for (int i = 1; i < C_CLS; ++i) mx = fmaxf(mx, logits[i]);
        float se = 0.0f;
        for (int i = 0; i < C_CLS; ++i) se += expf(logits[i] - mx);
        const float lse = logf(se) + mx;
        for (int i = 0; i < C_CLS; ++i) out[g * C_CLS + i] = logits[i] - lse;
    }
}

// =====================================================================
extern "C" void kernel_launch(void* const* d_in, const int* in_sizes, int n_in,
                              void* d_out, int out_size, void* d_ws, size_t ws_size,
                              hipStream_t stream)
{
    (void)in_sizes; (void)n_in; (void)out_size; (void)ws_size;

    const float* x     = (const float*)d_in[0];
    const int*   eidx  = (const int*)  d_in[1];   // [2, N_EDGES]: row0=src, row1=dst
    const int*   batch = (const int*)  d_in[2];
    const float* W1a = (const float*)d_in[3];  const float* b1a = (const float*)d_in[4];
    const float* g1a = (const float*)d_in[5];  const float* be1a= (const float*)d_in[6];
    const float* W2a = (const float*)d_in[7];  const float* b2a = (const float*)d_in[8];
    const float* g2a = (const float*)d_in[9];  const float* be2a= (const float*)d_in[10];
    const float* W1b = (const float*)d_in[11]; const float* b1b = (const float*)d_in[12];
    const float* g1b = (const float*)d_in[13]; const float* be1b= (const float*)d_in[14];
    const float* W2b = (const float*)d_in[15]; const float* b2b = (const float*)d_in[16];
    const float* g2b = (const float*)d_in[17]; const float* be2b= (const float*)d_in[18];
    const float* Wl0 = (const float*)d_in[19]; const float* bl0 = (const float*)d_in[20];
    const float* Wl1 = (const float*)d_in[21]; const float* bl1 = (const float*)d_in[22];

    // ---- workspace layout (floats) ----
    float* ws    = (float*)d_ws;
    float* stats = ws;                          // 4 layers x (sum[128] | sqs[128]) = 1024
    float* pool1 = ws + 1024;                   // 512*128
    float* pool2 = pool1 + N_GRAPHS * 128;      // 512*128
    float* cnt   = pool2 + N_GRAPHS * 128;      // 512
    float* scsh  = cnt + N_GRAPHS;              // 4 layers x (sc[128] | sh[128]) = 1024
    float* bufA  = scsh + 1024;                 // [N,128]
    float* bufB  = bufA + (size_t)N_NODES * 128;// [N,128]

    // zero all accumulators (stats + pools + cnt) -- graph-capture safe
    hipMemsetAsync(ws, 0, (size_t)(1024 + 2 * N_GRAPHS * 128 + N_GRAPHS) * sizeof(float),
                   stream);

    const int NT   = N_NODES / 16;              // 6250 row tiles
    const int GB   = 2048;                      // persistent GEMM blocks
    const int NE_B = (N_EDGES + 7) / 8;         // 8 edge-waves per 256-thread block
    const int EW_B = (N_NODES * 128) / 256;     // elementwise blocks
    const float invN = 1.0f / (float)N_NODES;

    // --- layer 0 MLP: z1 = x@W1a+b1a ; z2 = BNReLU(z1)@W2a+b2a ---
    gin_gemm128<<<GB, 256, 0, stream>>>(x,    W1a, b1a, nullptr,  nullptr,  0,
                                        bufA, stats + 0,   stats + 128, NT);
    gin_stats <<<1, 128, 0, stream>>>(stats + 0,   stats + 128, g1a, be1a,
                                      scsh + 0,   scsh + 128, invN);
    gin_gemm128<<<GB, 256, 0, stream>>>(bufA, W2a, b2a, scsh + 0, scsh + 128, 1,
                                        bufB, stats + 256, stats + 384, NT);
    gin_stats <<<1, 128, 0, stream>>>(stats + 256, stats + 384, g2a, be2a,
                                      scsh + 256, scsh + 384, invN);

    // --- h = BNReLU(z2): materialize h (bufA), init agg=h (bufB in-place),
    //     pool1 += h per graph, node counts ---
    gin_bnrelu_pool<<<EW_B, 256, 0, stream>>>(bufB, scsh + 256, scsh + 384, batch,
                                              bufA, bufB, pool1, cnt);

    // --- GIN aggregation: agg[dst] += h[src] over 3.2M edges ---
    gin_scatter<<<NE_B, 256, 0, stream>>>(bufA, eidx, eidx + N_EDGES, bufB, N_EDGES);

    // --- layer 1 MLP on agg: z3 = agg@W1b+b1b ; z4 = BNReLU(z3)@W2b+b2b ---
    gin_gemm128<<<GB, 256, 0, stream>>>(bufB, W1b, b1b, nullptr,    nullptr,    0,
                                        bufA, stats + 512, stats + 640, NT);
    gin_stats <<<1, 128, 0, stream>>>(stats + 512, stats + 640, g1b, be1b,
                                      scsh + 512, scsh + 640, invN);
    gin_gemm128<<<GB, 256, 0, stream>>>(bufA, W2b, b2b, scsh + 512, scsh + 640, 1,
                                        bufB, stats + 768, stats + 896, NT);
    gin_stats <<<1, 128, 0, stream>>>(stats + 768, stats + 896, g2b, be2b,
                                      scsh + 768, scsh + 896, invN);

    // --- h2 = BNReLU(z4): pooling only (never materialized) ---
    gin_bnrelu_pool<<<EW_B, 256, 0, stream>>>(bufB, scsh + 768, scsh + 896, batch,
                                              nullptr, nullptr, pool2, nullptr);

    // --- classifier heads + log_softmax ---
    gin_heads<<<N_GRAPHS, 64, 0, stream>>>(pool1, pool2, cnt,
                                           Wl0, bl0, Wl1, bl1, (float*)d_out);
}